// ResNetGCN_70858370450156
// MI455X (gfx1250) — compile-verified
//
#include <hip/hip_runtime.h>

typedef __attribute__((ext_vector_type(2))) float v2f;
typedef __attribute__((ext_vector_type(8))) float v8f;

constexpr int N = 40000, E = 640000, F = 512, H = 128, G = 64, C = 2;
constexpr int CAP = 128;              // per-node in-edge bucket capacity
constexpr float EPS = 1e-5f;

__global__ void k_zero(float* __restrict__ p, int n) {
  int i = blockIdx.x * blockDim.x + threadIdx.x;
  if (i < n) p[i] = 0.0f;
}

__global__ void k_zero_i(int* __restrict__ p, int n) {
  int i = blockIdx.x * blockDim.x + threadIdx.x;
  if (i < n) p[i] = 0;
}

// Counting-sort bucket build: bucket[d][pos] = edge id (int atomics only, once per call).
__global__ void k_bucket(const int* __restrict__ dst, int* __restrict__ cursor,
                         int* __restrict__ bucket) {
  int e = blockIdx.x * blockDim.x + threadIdx.x;
  if (e < E) {
    int d = dst[e];
    int pos = atomicAdd(&cursor[d], 1);
    if (pos < CAP) bucket[(long)d * CAP + pos] = e;
  }
}

__global__ void k_dinv(const int* __restrict__ cursor, float* __restrict__ dinv) {
  int i = blockIdx.x * blockDim.x + threadIdx.x;
  if (i < N) dinv[i] = rsqrtf((float)cursor[i] + 1.0f);
}

__global__ void k_norm(const int* __restrict__ src, const int* __restrict__ dst,
                       const float* __restrict__ dinv, float* __restrict__ norm_e) {
  int e = blockIdx.x * blockDim.x + threadIdx.x;
  if (e < E) norm_e[e] = dinv[src[e]] * dinv[dst[e]];
}

#define WMMA_F32(a, b, c) \
  __builtin_amdgcn_wmma_f32_16x16x4_f32(false, (a), false, (b), (short)0, (c), false, false)

// hw(N x Hc) = A(N x K) @ B(K x Hc), row-major.
// Block = 128 threads = 4 waves; each wave: 16 rows x 64 cols (4 f32-WMMA accumulators).
// Grid (Hc/64, N/64); dims divide exactly (N=40000, Hc=128, K in {512,128}).
__global__ void k_gemm_wmma(const float* __restrict__ A, const float* __restrict__ B,
                            float* __restrict__ hw, int K, int Hc) {
  const int lane = threadIdx.x & 31;
  const int wave = threadIdx.x >> 5;
  const int m0 = (blockIdx.y * 4 + wave) * 16;
  const int n0 = blockIdx.x * 64;
  const int half = lane >> 4;                    // 0: K=0..1 | 1: K=2..3
  const int l16  = lane & 15;
  const int col  = n0 + l16;

  const float* Ap = A + (long)(m0 + l16) * K + half * 2;   // A 16x4 lane layout
  const float* Bp = B + (long)(half * 2) * Hc + col;       // B 4x16 lane layout

  v8f acc0 = {}, acc1 = {}, acc2 = {}, acc3 = {};
#pragma unroll 2
  for (int k0 = 0; k0 < K; k0 += 4) {
    v2f a, b0, b1, b2, b3;
    a.x  = Ap[0];       a.y  = Ap[1];
    b0.x = Bp[0];       b0.y = Bp[Hc];
    b1.x = Bp[16];      b1.y = Bp[Hc + 16];
    b2.x = Bp[32];      b2.y = Bp[Hc + 32];
    b3.x = Bp[48];      b3.y = Bp[Hc + 48];
    acc0 = WMMA_F32(a, b0, acc0);
    acc1 = WMMA_F32(a, b1, acc1);
    acc2 = WMMA_F32(a, b2, acc2);
    acc3 = WMMA_F32(a, b3, acc3);
    Ap += 4;
    Bp += 4 * Hc;
  }
#pragma unroll
  for (int r = 0; r < 8; ++r) {
    const long row = m0 + half * 8 + r;
    float* hwp = hw + row * Hc + col;
    hwp[0]  = acc0[r];
    hwp[16] = acc1[r];
    hwp[32] = acc2[r];
    hwp[48] = acc3[r];
  }
}

// One wave per destination node; lane owns features 4*lane..4*lane+3.
// agg = sum_{in-edges} hw[src]*norm + hw[self]*dinv^2 + bias; then ReLU + eval-BN -> h.
// No float atomics: aggregation is register-resident, reads are coalesced 512B rows.
__global__ void k_gather(const float* __restrict__ hw, const int* __restrict__ src,
                         const float* __restrict__ norm_e,
                         const int* __restrict__ bucket, const int* __restrict__ cursor,
                         const float* __restrict__ dinv, const float* __restrict__ bias,
                         const float* __restrict__ g, const float* __restrict__ be,
                         const float* __restrict__ m, const float* __restrict__ v,
                         float* __restrict__ h) {
  const int node = (blockIdx.x * blockDim.x + threadIdx.x) >> 5;
  const int lane = threadIdx.x & 31;
  if (node >= N) return;

  // self-loop term
  float d = dinv[node];
  float sn = d * d;
  float4 acc = ((const float4*)(hw + (long)node * H))[lane];
  acc.x *= sn; acc.y *= sn; acc.z *= sn; acc.w *= sn;

  int cnt = cursor[node];
  if (cnt > CAP) cnt = CAP;
  const int* bk = bucket + (long)node * CAP;
  for (int j = 0; j < cnt; ++j) {
    int e = bk[j];                                   // wave-uniform
    float ne = norm_e[e];
    float4 r = ((const float4*)(hw + (long)src[e] * H))[lane];
    acc.x += r.x * ne; acc.y += r.y * ne; acc.z += r.z * ne; acc.w += r.w * ne;
  }

  const int f = lane * 4;
  const float4 b4  = *(const float4*)(bias + f);
  const float4 g4  = *(const float4*)(g + f);
  const float4 be4 = *(const float4*)(be + f);
  const float4 m4  = *(const float4*)(m + f);
  const float4 v4  = *(const float4*)(v + f);

  float4 o;
  o.x = (fmaxf(acc.x + b4.x, 0.0f) - m4.x) * rsqrtf(v4.x + EPS) * g4.x + be4.x;
  o.y = (fmaxf(acc.y + b4.y, 0.0f) - m4.y) * rsqrtf(v4.y + EPS) * g4.y + be4.y;
  o.z = (fmaxf(acc.z + b4.z, 0.0f) - m4.z) * rsqrtf(v4.z + EPS) * g4.z + be4.z;
  o.w = (fmaxf(acc.w + b4.w, 0.0f) - m4.w) * rsqrtf(v4.w + EPS) * g4.w + be4.w;
  ((float4*)(h + (long)node * H))[lane] = o;
}

__global__ void k_pool(const float* __restrict__ h, const int* __restrict__ batch,
                       float* __restrict__ pooled, float* __restrict__ cnt) {
  int i = blockIdx.x * blockDim.x + threadIdx.x;
  if (i < N * H) {
    int n = i >> 7, f = i & (H - 1);
    int gi = batch[n];
    atomicAdd(&pooled[gi * H + f], h[i]);
    if (f == 0) atomicAdd(&cnt[gi], 1.0f);
  }
}

__global__ void k_head(const float* __restrict__ pooled, const float* __restrict__ cnt,
                       const float* __restrict__ Wc, const float* __restrict__ bc,
                       float* __restrict__ out) {
  int t = threadIdx.x;            // 0..127
  int gi = t >> 1, c = t & 1;
  float inv = 1.0f / fmaxf(cnt[gi], 1.0f);
  float s = 0.0f;
#pragma unroll 8
  for (int f = 0; f < H; ++f) s += pooled[gi * H + f] * Wc[f * C + c];
  out[t] = s * inv + bc[c];
}

extern "C" void kernel_launch(void* const* d_in, const int* in_sizes, int n_in,
                              void* d_out, int out_size, void* d_ws, size_t ws_size,
                              hipStream_t stream) {
  (void)in_sizes; (void)n_in; (void)out_size; (void)ws_size;
  const float* x     = (const float*)d_in[0];
  const int*   ei    = (const int*)d_in[1];
  const int*   batch = (const int*)d_in[2];
  const float* W[3]  = {(const float*)d_in[3],  (const float*)d_in[9],  (const float*)d_in[15]};
  const float* bb[3] = {(const float*)d_in[4],  (const float*)d_in[10], (const float*)d_in[16]};
  const float* gg[3] = {(const float*)d_in[5],  (const float*)d_in[11], (const float*)d_in[17]};
  const float* bee[3]= {(const float*)d_in[6],  (const float*)d_in[12], (const float*)d_in[18]};
  const float* mm[3] = {(const float*)d_in[7],  (const float*)d_in[13], (const float*)d_in[19]};
  const float* vv[3] = {(const float*)d_in[8],  (const float*)d_in[14], (const float*)d_in[20]};
  const float* Wc = (const float*)d_in[21];
  const float* bc = (const float*)d_in[22];
  float* out = (float*)d_out;

  // workspace: dinv | norm_e | hw | h | pooled | cnt | cursor(int) | bucket(int)
  float* base   = (float*)d_ws;
  float* dinv   = base;                      // N
  float* norm   = dinv + N;                  // E
  float* hw     = norm + E;                  // N*H
  float* hbuf   = hw + (long)N * H;          // N*H
  float* pooled = hbuf + (long)N * H;        // G*H
  float* cnt    = pooled + G * H;            // G
  int*   cursor = (int*)(cnt + G);           // N
  int*   bucket = cursor + N;                // N*CAP

  const int* srcp = ei;
  const int* dstp = ei + E;

  k_zero_i<<<(N + 255) / 256, 256, 0, stream>>>(cursor, N);
  k_zero<<<(G * H + G + 255) / 256, 256, 0, stream>>>(pooled, G * H + G);
  k_bucket<<<E / 256, 256, 0, stream>>>(dstp, cursor, bucket);
  k_dinv<<<(N + 255) / 256, 256, 0, stream>>>(cursor, dinv);
  k_norm<<<E / 256, 256, 0, stream>>>(srcp, dstp, dinv, norm);

  const float* hin = x;
  int K = F;
  for (int l = 0; l < 3; ++l) {
    dim3 grid(H / 64, N / 64);               // 2 x 625, 4 waves/block, 4 accs/wave
    k_gemm_wmma<<<grid, 128, 0, stream>>>(hin, W[l], hw, K, H);
    k_gather<<<(N * 32) / 256, 256, 0, stream>>>(hw, srcp, norm, bucket, cursor, dinv,
                                                 bb[l], gg[l], bee[l], mm[l], vv[l], hbuf);
    hin = hbuf;
    K = H;
  }
  k_pool<<<(N * H + 255) / 256, 256, 0, stream>>>(hbuf, batch, pooled, cnt);
  k_head<<<1, G * C, 0, stream>>>(pooled, cnt, Wc, bc, out);
}